// AttentionNet_35356170781128
// MI455X (gfx1250) — compile-verified
//
#include <hip/hip_runtime.h>
#include <hip/hip_bf16.h>

typedef __attribute__((ext_vector_type(2))) float v2f;
typedef __attribute__((ext_vector_type(8))) float v8f;
typedef __attribute__((ext_vector_type(4))) unsigned int u32x4;
typedef __attribute__((ext_vector_type(8))) int i32x8;
typedef __attribute__((ext_vector_type(4))) int i32x4;

#define SEQ   829
#define SP    498
#define SE    331
#define BATCH 16
#define DMODEL 200
#define RNNH  100
#define NHEAD 8
#define HDIM  32

// ---------------------------------------------------------------------------
// Generic batched GEMM on V_WMMA_F32_16X16X4_F32.
// 128 threads = 4 waves; wave w owns M-tile (blockIdx.x*4 + w).
// Batch z decomposes as (i0 = z/n1, i1 = z%n1) with per-operand strides.
//   A: row-major [M,K], row stride lda
//   B: TRANSB==0 -> [K,N] row stride ldb ; TRANSB==1 -> stored [N,K] row stride ldb
// C = relu?( alpha * A@B + bias )
// Interior tiles run an unrolled guard-free loop (EXEC all-1s, 4 WMMAs/iter);
// ragged edges (M or N not tile-aligned) fall back to a value-guarded loop.
// ---------------------------------------------------------------------------
template<int TRANSB>
__global__ __launch_bounds__(128)
void gemm_wmma_f32(const float* __restrict__ A, const float* __restrict__ Bm,
                   float* __restrict__ C, const float* __restrict__ bias,
                   int M, int N, int K,
                   long long lda, long long ldb, long long ldc,
                   long long sA0, long long sA1,
                   long long sB0, long long sB1,
                   long long sC0, long long sC1,
                   int n1, float alpha, int relu)
{
    const int wave = threadIdx.x >> 5;
    const int lane = threadIdx.x & 31;
    const int m0   = (blockIdx.x * 4 + wave) * 16;
    if (m0 >= M) return;                       // wave-uniform

    const int z  = blockIdx.z;
    const int i0 = z / n1;
    const int i1 = z % n1;
    const float* Ab = A  + (long long)i0 * sA0 + (long long)i1 * sA1;
    const float* Bb = Bm + (long long)i0 * sB0 + (long long)i1 * sB1;
    float*       Cb = C  + (long long)i0 * sC0 + (long long)i1 * sC1;

    const int n0   = blockIdx.y * 16;
    const int half = lane >> 4;                // 0: lanes 0-15, 1: lanes 16-31
    const int l16  = lane & 15;

    v8f acc = {};
    const bool interior = (m0 + 16 <= M) && (n0 + 16 <= N);  // wave-uniform

    if (interior) {
        const float* arow = Ab + (long long)(m0 + l16) * lda;
        const float* brow = TRANSB ? (Bb + (long long)(n0 + l16) * ldb)
                                   : (Bb + (n0 + l16));
        int k = 0;
        for (; k + 16 <= K; k += 16) {
            __builtin_prefetch(arow + k + 32, 0, 1);   // global_prefetch_b8
            #pragma unroll
            for (int kk = 0; kk < 16; kk += 4) {
                const int k0 = k + kk + half * 2;
                v2f av; av.x = arow[k0]; av.y = arow[k0 + 1];
                v2f bv;
                if (TRANSB) { bv.x = brow[k0]; bv.y = brow[k0 + 1]; }
                else        { bv.x = brow[(long long)k0 * ldb];
                              bv.y = brow[(long long)(k0 + 1) * ldb]; }
                acc = __builtin_amdgcn_wmma_f32_16x16x4_f32(
                          false, av, false, bv, (short)0, acc, false, false);
            }
        }
        for (; k < K; k += 4) {                 // K remainder (< 16), K-guards only
            const int k0  = k + half * 2;
            const int k0c = k0     < K ? k0     : 0;
            const int k1c = k0 + 1 < K ? k0 + 1 : 0;
            float a0 = arow[k0c], a1 = arow[k1c];
            float b0, b1;
            if (TRANSB) { b0 = brow[k0c]; b1 = brow[k1c]; }
            else        { b0 = brow[(long long)k0c * ldb];
                          b1 = brow[(long long)k1c * ldb]; }
            a0 = (k0     < K) ? a0 : 0.0f;  b0 = (k0     < K) ? b0 : 0.0f;
            a1 = (k0 + 1 < K) ? a1 : 0.0f;  b1 = (k0 + 1 < K) ? b1 : 0.0f;
            v2f av; av.x = a0; av.y = a1;
            v2f bv; bv.x = b0; bv.y = b1;
            acc = __builtin_amdgcn_wmma_f32_16x16x4_f32(
                      false, av, false, bv, (short)0, acc, false, false);
        }
    } else {
        // Ragged edge tile: clamp addresses, zero values by select.
        const int am   = m0 + l16;
        const int am_c = am < M ? am : (M - 1);
        const int bn   = n0 + l16;
        const int bn_c = bn < N ? bn : (N - 1);
        const float* arow = Ab + (long long)am_c * lda;
        for (int k = 0; k < K; k += 4) {
            const int k0  = k + half * 2;
            const int k0c = k0     < K ? k0     : 0;
            const int k1c = k0 + 1 < K ? k0 + 1 : 0;
            float a0 = arow[k0c];
            float a1 = arow[k1c];
            a0 = (am < M && k0     < K) ? a0 : 0.0f;
            a1 = (am < M && k0 + 1 < K) ? a1 : 0.0f;
            float b0, b1;
            if (TRANSB) { b0 = Bb[(long long)bn_c * ldb + k0c];
                          b1 = Bb[(long long)bn_c * ldb + k1c]; }
            else        { b0 = Bb[(long long)k0c * ldb + bn_c];
                          b1 = Bb[(long long)k1c * ldb + bn_c]; }
            b0 = (bn < N && k0     < K) ? b0 : 0.0f;
            b1 = (bn < N && k0 + 1 < K) ? b1 : 0.0f;
            v2f av; av.x = a0; av.y = a1;
            v2f bv; bv.x = b0; bv.y = b1;
            acc = __builtin_amdgcn_wmma_f32_16x16x4_f32(
                      false, av, false, bv, (short)0, acc, false, false);
        }
    }

    // C/D layout: VGPR r -> row m0 + (half ? r+8 : r), col n0 + l16
    #pragma unroll
    for (int r = 0; r < 8; ++r) {
        const int row = m0 + (half ? r + 8 : r);
        const int col = n0 + l16;
        if (row < M && col < N) {
            float v = acc[r] * alpha;
            if (bias) v += bias[col];
            if (relu) v = v > 0.0f ? v : 0.0f;
            Cb[(long long)row * ldc + col] = v;
        }
    }
}

// ---------------------------------------------------------------------------
// Conv1d(4->200, k=13, VALID) + BN(eval) + ReLU + MaxPool1d(6), both branches,
// writing time-major features feat[s][b][c]  (s: 0..497 = p, 498..828 = e).
// ---------------------------------------------------------------------------
__global__ void conv_bn_pool_kernel(
    const float* __restrict__ xp, const float* __restrict__ xe,
    const float* __restrict__ wp, const float* __restrict__ gp,
    const float* __restrict__ bp, const float* __restrict__ mp,
    const float* __restrict__ vp,
    const float* __restrict__ we, const float* __restrict__ ge,
    const float* __restrict__ be, const float* __restrict__ me,
    const float* __restrict__ ve,
    float* __restrict__ feat)
{
    const long long idx   = (long long)blockIdx.x * blockDim.x + threadIdx.x;
    const long long total = (long long)SEQ * BATCH * DMODEL;
    if (idx >= total) return;
    const int c = (int)(idx % DMODEL);
    const int b = (int)((idx / DMODEL) % BATCH);
    const int s = (int)(idx / (DMODEL * BATCH));

    const float* x; const float* w; float inv, shift; int L, sp;
    if (s < SP) {
        x = xp + (long long)b * 4 * 3000; w = wp + c * 52;
        inv = gp[c] * rsqrtf(vp[c] + 1e-5f); shift = bp[c] - mp[c] * inv;
        L = 3000; sp = s;
    } else {
        x = xe + (long long)b * 4 * 2000; w = we + c * 52;
        inv = ge[c] * rsqrtf(ve[c] + 1e-5f); shift = be[c] - me[c] * inv;
        L = 2000; sp = s - SP;
    }

    float best = 0.0f;   // post-ReLU values are >= 0
    for (int j = 0; j < 6; ++j) {
        const int pos = sp * 6 + j;
        float acc = 0.0f;
        #pragma unroll
        for (int i = 0; i < 4; ++i) {
            const float* xr = x + i * L + pos;
            const float* wr = w + i * 13;
            #pragma unroll
            for (int f = 0; f < 13; ++f) acc += xr[f] * wr[f];
        }
        float y = acc * inv + shift;
        y = y > 0.0f ? y : 0.0f;
        best = y > best ? y : best;
    }
    feat[((long long)s * BATCH + b) * DMODEL + c] = best;
}

// ---------------------------------------------------------------------------
// LSTM recurrence for one (layer, dir). Input projections precomputed by the
// WMMA GEMM: gates[t][b][400] (torch gate order i,f,g,o).
// w_hh (400x100 f32 = 160KB) is staged into LDS ONCE by the Tensor Data Mover
// (tensor_load_to_lds + s_wait_tensorcnt) -- CDNA5's 320KB LDS holds it, so
// the 829 sequential steps read weights at LDS latency instead of L2.
// Dynamic LDS layout: [ whh 40000 f32 | h double-buffer 2*1600 f32 ].
// ---------------------------------------------------------------------------
__global__ __launch_bounds__(512)
void lstm_scan_kernel(const float* __restrict__ gates,
                      const float* __restrict__ whh,   // [400][100] global
                      const float* __restrict__ bih,   // [400]
                      const float* __restrict__ bhh,   // [400]
                      float* __restrict__ outbuf,      // [S][B][200]
                      int S, int dir)
{
    extern __shared__ float smem[];
    float* whh_lds = smem;                 // 400*100
    float* hb      = smem + 400 * RNNH;    // 2 * BATCH*RNNH

    const int tid = threadIdx.x;

    // --- TDM: DMA w_hh global -> LDS (wave 0 issues; others wait at barrier)
    if (tid < 32) {
        const unsigned lds_off = (unsigned)(unsigned long long)(void*)whh_lds;
        const unsigned long long ga = (unsigned long long)whh;
        u32x4 g0;
        g0[0] = 1u;                                   // count=1 (valid D#)
        g0[1] = lds_off;                              // lds_addr (bytes)
        g0[2] = (unsigned)(ga & 0xFFFFFFFFu);         // global_addr[31:0]
        g0[3] = (unsigned)((ga >> 32) & 0x1FFFFFFu)   // global_addr[56:32]
              | (2u << 30);                           // type=2 ("image")
        i32x8 g1;
        g1[0] = (int)(2u << 16);                      // data_size=2 -> 4 bytes
        g1[1] = (int)((unsigned)RNNH << 16);          // tensor_dim0 = 100 (lo16)
        g1[2] = (int)((unsigned)400 << 16);           // dim0 hi=0 | tensor_dim1 = 400 (lo16)
        g1[3] = (int)((unsigned)RNNH << 16);          // dim1 hi=0 | tile_dim0 = 100
        g1[4] = (int)400;                             // tile_dim1 = 400 | tile_dim2 = 0
        g1[5] = (int)RNNH;                            // tensor_dim0_stride = 100 (lo32)
        g1[6] = 0;                                    // stride0 hi | stride1 lo
        g1[7] = 0;
        i32x4 g2 = {0, 0, 0, 0};
        i32x4 g3 = {0, 0, 0, 0};
#if defined(__clang_major__) && (__clang_major__ >= 23)
        i32x8 g4 = {0, 0, 0, 0, 0, 0, 0, 0};
        __builtin_amdgcn_tensor_load_to_lds(g0, g1, g2, g3, g4, 0);
#else
        __builtin_amdgcn_tensor_load_to_lds(g0, g1, g2, g3, 0);
#endif
        __builtin_amdgcn_s_wait_tensorcnt(0);
    }
    for (int i = tid; i < BATCH * RNNH; i += 512) {
        hb[i] = 0.0f; hb[BATCH * RNNH + i] = 0.0f;
    }
    float creg[4] = {0.0f, 0.0f, 0.0f, 0.0f};
    __syncthreads();

    int cur = 0;
    for (int step = 0; step < S; ++step) {
        const int t = dir ? (S - 1 - step) : step;
        const float* g_t = gates + (long long)t * BATCH * 400;
        for (int u = 0; u < 4; ++u) {
            const int unit = tid + u * 512;
            if (unit < BATCH * RNNH) {
                const int b = unit / RNNH, j = unit % RNNH;
                const float* h = &hb[cur * BATCH * RNNH + b * RNNH];
                float gi = g_t[b * 400 + j]       + bih[j]       + bhh[j];
                float gf = g_t[b * 400 + 100 + j] + bih[100 + j] + bhh[100 + j];
                float gg = g_t[b * 400 + 200 + j] + bih[200 + j] + bhh[200 + j];
                float go = g_t[b * 400 + 300 + j] + bih[300 + j] + bhh[300 + j];
                const float* wi = whh_lds + j * RNNH;
                const float* wf = whh_lds + (100 + j) * RNNH;
                const float* wg = whh_lds + (200 + j) * RNNH;
                const float* wo = whh_lds + (300 + j) * RNNH;
                for (int j2 = 0; j2 < RNNH; ++j2) {
                    const float hv = h[j2];
                    gi += hv * wi[j2]; gf += hv * wf[j2];
                    gg += hv * wg[j2]; go += hv * wo[j2];
                }
                const float si = 1.0f / (1.0f + __expf(-gi));
                const float sf = 1.0f / (1.0f + __expf(-gf));
                const float so = 1.0f / (1.0f + __expf(-go));
                const float tg = tanhf(gg);
                const float cc = sf * creg[u] + si * tg;
                creg[u] = cc;
                const float hn = so * tanhf(cc);
                hb[(1 - cur) * BATCH * RNNH + b * RNNH + j] = hn;
                outbuf[((long long)t * BATCH + b) * DMODEL + dir * RNNH + j] = hn;
            }
        }
        __syncthreads();
        cur = 1 - cur;
    }
}

// Pack wq/wk/wv [8,200,32] -> Wqkv [200,768] and bq/bk/bv -> bias [768]
__global__ void pack_qkv_kernel(
    const float* __restrict__ wq, const float* __restrict__ wk,
    const float* __restrict__ wv, const float* __restrict__ bq,
    const float* __restrict__ bk, const float* __restrict__ bv,
    float* __restrict__ W, float* __restrict__ Bz)
{
    const int idx = blockIdx.x * blockDim.x + threadIdx.x;
    if (idx < DMODEL * 768) {
        const int n = idx % 768, k = idx / 768;
        const int proj = n / 256, rem = n % 256, h = rem / 32, kk = rem % 32;
        const float* src = proj == 0 ? wq : (proj == 1 ? wk : wv);
        W[idx] = src[((long long)h * DMODEL + k) * HDIM + kk];
    }
    if (idx < 768) {
        const int proj = idx / 256, rem = idx % 256, h = rem / 32, kk = rem % 32;
        const float* sb = proj == 0 ? bq : (proj == 1 ? bk : bv);
        Bz[idx] = sb[h * HDIM + kk];
    }
}

// Row softmax, in place; one block per row (rows contiguous, len elems).
__global__ __launch_bounds__(256)
void softmax_rows_kernel(float* __restrict__ data, int len)
{
    __shared__ float red[256];
    float* p = data + (long long)blockIdx.x * len;
    const int tid = threadIdx.x;

    float mx = -3.4e38f;
    for (int t = tid; t < len; t += 256) mx = fmaxf(mx, p[t]);
    red[tid] = mx; __syncthreads();
    for (int s = 128; s > 0; s >>= 1) {
        if (tid < s) red[tid] = fmaxf(red[tid], red[tid + s]);
        __syncthreads();
    }
    mx = red[0]; __syncthreads();

    float sum = 0.0f;
    for (int t = tid; t < len; t += 256) {
        const float e = __expf(p[t] - mx);
        p[t] = e; sum += e;
    }
    red[tid] = sum; __syncthreads();
    for (int s = 128; s > 0; s >>= 1) {
        if (tid < s) red[tid] += red[tid + s];
        __syncthreads();
    }
    const float inv = 1.0f / red[0];
    for (int t = tid; t < len; t += 256) p[t] *= inv;
}

// Sum over S, global mean/std (ddof=1) normalize, fc3 + softmax -> probs[16,2]
__global__ __launch_bounds__(256)
void final_head_kernel(const float* __restrict__ hbufg,  // [B*S][100]
                       const float* __restrict__ fc3w,   // [100][2]
                       const float* __restrict__ fc3b,   // [2]
                       float* __restrict__ probs)        // [16*2]
{
    __shared__ float r[BATCH * RNNH];
    __shared__ float red[256];
    const int tid = threadIdx.x;

    for (int u = tid; u < BATCH * RNNH; u += 256) {
        const int b = u / RNNH, j = u % RNNH;
        float acc = 0.0f;
        const float* col = hbufg + (long long)b * SEQ * RNNH + j;
        for (int s = 0; s < SEQ; ++s) acc += col[(long long)s * RNNH];
        r[u] = acc;
    }
    __syncthreads();

    float lsum = 0.0f;
    for (int u = tid; u < BATCH * RNNH; u += 256) lsum += r[u];
    red[tid] = lsum; __syncthreads();
    for (int s = 128; s > 0; s >>= 1) { if (tid < s) red[tid] += red[tid + s]; __syncthreads(); }
    const float mean = red[0] / (float)(BATCH * RNNH);
    __syncthreads();

    float lv = 0.0f;
    for (int u = tid; u < BATCH * RNNH; u += 256) { const float d = r[u] - mean; lv += d * d; }
    red[tid] = lv; __syncthreads();
    for (int s = 128; s > 0; s >>= 1) { if (tid < s) red[tid] += red[tid + s]; __syncthreads(); }
    const float stdv = sqrtf(red[0] / (float)(BATCH * RNNH - 1));
    __syncthreads();

    for (int u = tid; u < BATCH * RNNH; u += 256) r[u] = (r[u] - mean) / stdv;
    __syncthreads();

    if (tid < BATCH) {
        const int b = tid;
        float l0 = fc3b[0], l1 = fc3b[1];
        for (int j = 0; j < RNNH; ++j) {
            const float v = r[b * RNNH + j];
            l0 += v * fc3w[j * 2]; l1 += v * fc3w[j * 2 + 1];
        }
        const float m = fmaxf(l0, l1);
        const float e0 = __expf(l0 - m), e1 = __expf(l1 - m);
        const float inv = 1.0f / (e0 + e1);
        probs[b * 2]     = e0 * inv;
        probs[b * 2 + 1] = e1 * inv;
    }
}

// ---------------------------------------------------------------------------
extern "C" void kernel_launch(void* const* d_in, const int* in_sizes, int n_in,
                              void* d_out, int out_size, void* d_ws, size_t ws_size,
                              hipStream_t stream) {
    (void)in_sizes; (void)n_in; (void)out_size; (void)ws_size;

    const float* input_p  = (const float*)d_in[0];
    const float* input_e  = (const float*)d_in[1];
    const float* conv_p_w = (const float*)d_in[2];
    const float* bn_p_g   = (const float*)d_in[3];
    const float* bn_p_b   = (const float*)d_in[4];
    const float* bn_p_m   = (const float*)d_in[5];
    const float* bn_p_v   = (const float*)d_in[6];
    const float* conv_e_w = (const float*)d_in[7];
    const float* bn_e_g   = (const float*)d_in[8];
    const float* bn_e_b   = (const float*)d_in[9];
    const float* bn_e_m   = (const float*)d_in[10];
    const float* bn_e_v   = (const float*)d_in[11];
    const float* w_ih     = (const float*)d_in[12];  // [2,2,400,200]
    const float* w_hh     = (const float*)d_in[13];  // [2,2,400,100]
    const float* b_ih     = (const float*)d_in[14];  // [2,2,400]
    const float* b_hh     = (const float*)d_in[15];  // [2,2,400]
    const float* wq       = (const float*)d_in[16];
    const float* bq       = (const float*)d_in[17];
    const float* wk       = (const float*)d_in[18];
    const float* bk       = (const float*)d_in[19];
    const float* wv       = (const float*)d_in[20];
    const float* bv       = (const float*)d_in[21];
    const float* mh_w     = (const float*)d_in[22];  // [256,100]
    const float* mh_b     = (const float*)d_in[23];
    const float* fc3_w    = (const float*)d_in[24];  // [100,2]
    const float* fc3_b    = (const float*)d_in[25];

    float* out   = (float*)d_out;
    float* pattn = out + 32;                       // [B][S][H][S] flat

    // workspace layout (floats)
    float* ws    = (float*)d_ws;
    const long long nFeat = (long long)SEQ * BATCH * DMODEL;    // 2,652,800
    float* feat  = ws;
    float* lstmA = feat  + nFeat;
    float* lstmB = lstmA + nFeat;
    float* gates = lstmB + nFeat;                  // [2][S*B][400]
    float* wqkv  = gates + 2LL * SEQ * BATCH * 400;
    float* bqkv  = wqkv  + (long long)DMODEL * 768;
    float* qkv   = bqkv  + 768;                    // [S*B][768]
    float* attnc = qkv   + (long long)SEQ * BATCH * 768;  // [B*S][256]
    float* hout  = attnc + (long long)BATCH * SEQ * 256;  // [B*S][100]

    const long long MSB = (long long)SEQ * BATCH;  // 13264 (multiple of 16)
    const size_t lstm_smem = (size_t)(400 * RNNH + 2 * BATCH * RNNH) * sizeof(float);

    // 1) conv + BN + ReLU + pool -> feat[s][b][200]
    {
        const long long total = (long long)SEQ * BATCH * DMODEL;
        conv_bn_pool_kernel<<<dim3((unsigned)((total + 255) / 256)), dim3(256), 0, stream>>>(
            input_p, input_e, conv_p_w, bn_p_g, bn_p_b, bn_p_m, bn_p_v,
            conv_e_w, bn_e_g, bn_e_b, bn_e_m, bn_e_v, feat);
    }

    // 2) biLSTM: two layers; input projection via WMMA, recurrence via TDM+LDS scan
    for (int layer = 0; layer < 2; ++layer) {
        const float* in  = layer == 0 ? feat  : lstmA;
        float*       lo  = layer == 0 ? lstmA : lstmB;
        // gates[dir][t*B+b][400] = in @ w_ih[layer][dir]^T   (batched over dir)
        gemm_wmma_f32<1><<<dim3(208, 25, 2), dim3(128), 0, stream>>>(
            in, w_ih + (long long)layer * 2 * 400 * DMODEL, gates, nullptr,
            (int)MSB, 400, DMODEL,
            DMODEL, DMODEL, 400,
            0, 0,
            400LL * DMODEL, 0,
            MSB * 400, 0,
            1, 1.0f, 0);
        for (int dir = 0; dir < 2; ++dir) {
            lstm_scan_kernel<<<dim3(1), dim3(512), lstm_smem, stream>>>(
                gates + (long long)dir * MSB * 400,
                w_hh + ((long long)layer * 2 + dir) * 400 * RNNH,
                b_ih + ((long long)layer * 2 + dir) * 400,
                b_hh + ((long long)layer * 2 + dir) * 400,
                lo, SEQ, dir);
        }
    }

    // 3) pack QKV weights -> [200,768] + bias[768]; project with WMMA
    pack_qkv_kernel<<<dim3((DMODEL * 768 + 255) / 256), dim3(256), 0, stream>>>(
        wq, wk, wv, bq, bk, bv, wqkv, bqkv);
    gemm_wmma_f32<0><<<dim3(208, 48, 1), dim3(128), 0, stream>>>(
        lstmB, wqkv, qkv, bqkv,
        (int)MSB, 768, DMODEL,
        DMODEL, 768, 768,
        0, 0, 0, 0, 0, 0,
        1, 1.0f, 0);

    // 4) scores = Q @ K^T / sqrt(32), written straight into d_out in the
    //    final [B][S][H][S] layout; batched over (b,h) -> z = b*8 + h
    gemm_wmma_f32<1><<<dim3(13, 52, 128), dim3(128), 0, stream>>>(
        qkv, qkv + 256, pattn, nullptr,
        SEQ, SEQ, HDIM,
        (long long)BATCH * 768, (long long)BATCH * 768, (long long)NHEAD * SEQ,
        768, HDIM,
        768, HDIM,
        (long long)SEQ * NHEAD * SEQ, SEQ,
        NHEAD, 0.17677669529663687f, 0);

    // 5) softmax over last dim, in place (rows contiguous: B*S*H rows)
    softmax_rows_kernel<<<dim3(BATCH * SEQ * NHEAD), dim3(256), 0, stream>>>(pattn, SEQ);

    // 6) attn = relu(P @ V) into head-major concat [B*S][256]
    gemm_wmma_f32<0><<<dim3(13, 2, 128), dim3(128), 0, stream>>>(
        pattn, qkv + 512, attnc, nullptr,
        SEQ, HDIM, SEQ,
        (long long)NHEAD * SEQ, (long long)BATCH * 768, 256,
        (long long)SEQ * NHEAD * SEQ, SEQ,
        768, HDIM,
        (long long)SEQ * 256, HDIM,
        NHEAD, 1.0f, 1);

    // 7) h = relu(attn_cat @ mh_w + mh_b)  -> [B*S][100]
    gemm_wmma_f32<0><<<dim3(208, 7, 1), dim3(128), 0, stream>>>(
        attnc, mh_w, hout, mh_b,
        (int)MSB, RNNH, 256,
        256, RNNH, RNNH,
        0, 0, 0, 0, 0, 0,
        1, 1.0f, 1);

    // 8) sum over S, normalize, fc3 softmax -> probs
    final_head_kernel<<<dim3(1), dim3(256), 0, stream>>>(hout, fc3_w, fc3_b, out);
}